// Embedding_48713519071467
// MI455X (gfx1250) — compile-verified
//
#include <hip/hip_runtime.h>

typedef __attribute__((ext_vector_type(2))) float v2f;
typedef __attribute__((ext_vector_type(8))) float v8f;

static constexpr int B_  = 8;
static constexpr int N_  = 2048;
static constexpr int KNN = 32;

// ---------------------------------------------------------------------------
// f32 WMMA: D(16x16) = A(16x4) * B(4x16) + C   (CDNA5 V_WMMA_F32_16X16X4_F32)
// A frag (2 VGPR): lane l -> row m=l&15, a[v] = A[m][2*(l>>4)+v]
// B frag (2 VGPR): lane l -> col n=l&15, b[v] = B[2*(l>>4)+v][n]
// C/D  (8 VGPR):   lane l -> col n=l&15, c[v] = C[v + 8*(l>>4)][n]
// ---------------------------------------------------------------------------
__device__ __forceinline__ v8f wmma4f32(v2f a, v2f b, v8f c) {
#if __has_builtin(__builtin_amdgcn_wmma_f32_16x16x4_f32)
  return __builtin_amdgcn_wmma_f32_16x16x4_f32(false, a, false, b, (short)0, c,
                                               false, false);
#else
#warning "f32 16x16x4 wmma builtin unavailable on this toolchain"
  return c;
#endif
}

// ---------------------------------------------------------------------------
// x (B,3,N) -> p (B,N,3)
// ---------------------------------------------------------------------------
__global__ void xpose_kernel(const float* __restrict__ x, float* __restrict__ p) {
  const int i = blockIdx.x * blockDim.x + threadIdx.x;   // B_*N_ threads
  const int b = i / N_;
  const int n = i % N_;
#pragma unroll
  for (int c = 0; c < 3; ++c)
    p[(size_t)i * 3 + c] = x[((size_t)b * 3 + c) * N_ + n];
}

// ---------------------------------------------------------------------------
// top-32 of d = 2 p.q - |p|^2 - |q|^2 over 3-dim points (one thread / point)
// ---------------------------------------------------------------------------
__launch_bounds__(128)
__global__ void topk_xyz_kernel(const float* __restrict__ p, int* __restrict__ nbr) {
  __shared__ float sp[128 * 3];
  const int tid = threadIdx.x;
  const int b   = blockIdx.x >> 4;                 // 16 blocks per batch
  const int n   = ((blockIdx.x & 15) << 7) + tid;
  const float px = p[((size_t)b * N_ + n) * 3 + 0];
  const float py = p[((size_t)b * N_ + n) * 3 + 1];
  const float pz = p[((size_t)b * N_ + n) * 3 + 2];
  const float sqn = px * px + py * py + pz * pz;

  float bd[KNN];
  int   bi[KNN];
#pragma unroll
  for (int j = 0; j < KNN; ++j) { bd[j] = -3.0e38f; bi[j] = 0; }

  for (int ch = 0; ch < N_ / 128; ++ch) {
    __syncthreads();
    for (int i = tid; i < 128 * 3; i += 128)
      sp[i] = p[((size_t)b * N_ + ch * 128) * 3 + i];
    __syncthreads();
    for (int j = 0; j < 128; ++j) {
      const float qx = sp[j * 3 + 0];
      const float qy = sp[j * 3 + 1];
      const float qz = sp[j * 3 + 2];
      const float d  = 2.f * (px * qx + py * qy + pz * qz) - sqn
                       - (qx * qx + qy * qy + qz * qz);
      if (d > bd[KNN - 1]) {
        float cd = d; int ci = ch * 128 + j;
#pragma unroll
        for (int t = 0; t < KNN; ++t) {
          const bool g = cd > bd[t];
          const float td = bd[t]; const int ti = bi[t];
          bd[t] = g ? cd : td;  bi[t] = g ? ci : ti;
          cd    = g ? td : cd;  ci    = g ? ti : ci;
        }
      }
    }
  }
#pragma unroll
  for (int t = 0; t < KNN; ++t)
    nbr[((size_t)b * N_ + n) * KNN + t] = bi[t];
}

// ---------------------------------------------------------------------------
// squared norms of 64-dim features
// ---------------------------------------------------------------------------
__global__ void sqnorm64_kernel(const float* __restrict__ f, float* __restrict__ sq) {
  const int i = blockIdx.x * blockDim.x + threadIdx.x;   // B_*N_ threads
  float s = 0.f;
#pragma unroll
  for (int c = 0; c < 64; ++c) { const float v = f[(size_t)i * 64 + c]; s += v * v; }
  sq[i] = s;
}

// ---------------------------------------------------------------------------
// top-32 over 64-dim distances (one thread / point, LDS-chunked candidates)
// ---------------------------------------------------------------------------
__launch_bounds__(128)
__global__ void topk_feat_kernel(const float* __restrict__ f,
                                 const float* __restrict__ sq,
                                 int* __restrict__ nbr) {
  __shared__ float sf[64 * 64];
  __shared__ float ssq[64];
  const int tid = threadIdx.x;
  const int b   = blockIdx.x >> 4;
  const int n   = ((blockIdx.x & 15) << 7) + tid;

  float fe[64];
#pragma unroll
  for (int c = 0; c < 64; ++c) fe[c] = f[((size_t)b * N_ + n) * 64 + c];
  const float sqn = sq[(size_t)b * N_ + n];

  float bd[KNN];
  int   bi[KNN];
#pragma unroll
  for (int j = 0; j < KNN; ++j) { bd[j] = -3.0e38f; bi[j] = 0; }

  for (int ch = 0; ch < N_ / 64; ++ch) {
    __syncthreads();
    for (int i = tid; i < 64 * 64; i += 128)
      sf[i] = f[((size_t)b * N_ + ch * 64) * 64 + i];
    if (tid < 64) ssq[tid] = sq[(size_t)b * N_ + ch * 64 + tid];
    __syncthreads();
    for (int j = 0; j < 64; ++j) {
      float dot = 0.f;
#pragma unroll
      for (int c = 0; c < 64; ++c) dot += fe[c] * sf[j * 64 + c];
      const float d = 2.f * dot - sqn - ssq[j];
      if (d > bd[KNN - 1]) {
        float cd = d; int ci = ch * 64 + j;
#pragma unroll
        for (int t = 0; t < KNN; ++t) {
          const bool g = cd > bd[t];
          const float td = bd[t]; const int ti = bi[t];
          bd[t] = g ? cd : td;  bi[t] = g ? ci : ti;
          cd    = g ? td : cd;  ci    = g ? ti : ci;
        }
      }
    }
  }
#pragma unroll
  for (int t = 0; t < KNN; ++t)
    nbr[((size_t)b * N_ + n) * KNN + t] = bi[t];
}

// ---------------------------------------------------------------------------
// EdgeConv MLP: gather edge features, (2C->O1) GEMM + BN/LReLU,
// (O1->O2) GEMM + BN/LReLU, max over K. One point per wave, 4 waves per WG.
// All GEMMs via v_wmma_f32_16x16x4_f32.
// ---------------------------------------------------------------------------
template <int C, int FP, int O1, int O2>
__launch_bounds__(128, 1)
__global__ void edge_mlp_kernel(const float* __restrict__ fin,
                                const int*   __restrict__ nbr,
                                const float* __restrict__ wA,
                                const float* __restrict__ gA, const float* __restrict__ bA,
                                const float* __restrict__ mA, const float* __restrict__ vA,
                                const float* __restrict__ wB,
                                const float* __restrict__ gB, const float* __restrict__ bB,
                                const float* __restrict__ mB, const float* __restrict__ vB,
                                float* __restrict__ fout) {
  extern __shared__ float smem[];
  constexpr int NW = 4;
  float* sW1 = smem;                    // [FP][O1]   w_a^T, K-padded
  float* sW2 = sW1 + FP * O1;           // [O1][O2]   w_b^T
  float* sS1 = sW2 + O1 * O2;           // scale1 [O1]
  float* sB1 = sS1 + O1;                // shift1 [O1]
  float* sS2 = sB1 + O1;                // scale2 [O2]
  float* sB2 = sS2 + O2;                // shift2 [O2]
  float* sFeatBase = sB2 + O2;

  const int tid  = threadIdx.x;
  const int wave = tid >> 5;
  const int lane = tid & 31;
  float* sFeat = sFeatBase + wave * (32 * FP);                 // [32][FP]
  float* sHb   = sFeatBase + NW * (32 * FP) + wave * (32 * O1); // [32][O1]

  // ---- stage weights + fused BN scale/shift ----
  for (int i = tid; i < FP * O1; i += 128) {
    const int k = i / O1, o = i % O1;
    sW1[i] = (k < 2 * C) ? wA[o * (2 * C) + k] : 0.0f;
  }
  for (int i = tid; i < O1 * O2; i += 128) {
    const int k = i / O2, o = i % O2;
    sW2[i] = wB[o * O1 + k];
  }
  for (int i = tid; i < O1; i += 128) {
    const float s = gA[i] * rsqrtf(vA[i] + 1e-5f);
    sS1[i] = s; sB1[i] = bA[i] - mA[i] * s;
  }
  if (tid < O2) {
    const float s = gB[tid] * rsqrtf(vB[tid] + 1e-5f);
    sS2[tid] = s; sB2[tid] = bB[tid] - mB[tid] * s;
  }
  __syncthreads();

  const int pt = blockIdx.x * NW + wave;   // global point id, < B_*N_
  const int bb = pt / N_;

  // ---- build 32 x FP edge-feature tile (row = neighbor slot = lane) ----
  {
    const int ii = nbr[(size_t)pt * KNN + lane];
    const float* ctr = fin + (size_t)pt * C;
    const float* nb  = fin + ((size_t)bb * N_ + ii) * C;
#pragma unroll
    for (int c = 0; c < C; ++c) {
      const float cv = ctr[c];
      sFeat[lane * FP + c]     = cv;
      sFeat[lane * FP + C + c] = nb[c] - cv;
    }
#pragma unroll
    for (int c = 2 * C; c < FP; ++c) sFeat[lane * FP + c] = 0.0f;
  }
  __syncthreads();

  const int cl = lane & 15;
  const int kh = lane >> 4;

  // ---- GEMM1: (32 x FP) @ (FP x O1), BN + LeakyReLU -> sHb ----
#pragma unroll 1
  for (int nt = 0; nt < O1 / 16; ++nt) {
    v8f acc0 = {0.f,0.f,0.f,0.f,0.f,0.f,0.f,0.f};
    v8f acc1 = {0.f,0.f,0.f,0.f,0.f,0.f,0.f,0.f};
    const int col = nt * 16 + cl;
    for (int kt = 0; kt < FP / 4; ++kt) {
      const int kb = kt * 4 + 2 * kh;
      const v2f a0 = *(const v2f*)(sFeat + cl * FP + kb);
      const v2f a1 = *(const v2f*)(sFeat + (16 + cl) * FP + kb);
      v2f bf;
      bf.x = sW1[kb * O1 + col];
      bf.y = sW1[(kb + 1) * O1 + col];
      acc0 = wmma4f32(a0, bf, acc0);
      acc1 = wmma4f32(a1, bf, acc1);
    }
    const float s = sS1[col], sh = sB1[col];
#pragma unroll
    for (int v = 0; v < 8; ++v) {
      const int r = v + 8 * kh;
      float h0 = acc0[v] * s + sh; h0 = (h0 > 0.f) ? h0 : 0.2f * h0;
      float h1 = acc1[v] * s + sh; h1 = (h1 > 0.f) ? h1 : 0.2f * h1;
      sHb[r * O1 + col]        = h0;
      sHb[(16 + r) * O1 + col] = h1;
    }
  }
  __syncthreads();

  // ---- GEMM2: (32 x O1) @ (O1 x O2), BN + LeakyReLU, max over 32 rows ----
#pragma unroll 1
  for (int nt = 0; nt < O2 / 16; ++nt) {
    v8f acc0 = {0.f,0.f,0.f,0.f,0.f,0.f,0.f,0.f};
    v8f acc1 = {0.f,0.f,0.f,0.f,0.f,0.f,0.f,0.f};
    const int col = nt * 16 + cl;
    for (int kt = 0; kt < O1 / 4; ++kt) {
      const int kb = kt * 4 + 2 * kh;
      const v2f a0 = *(const v2f*)(sHb + cl * O1 + kb);
      const v2f a1 = *(const v2f*)(sHb + (16 + cl) * O1 + kb);
      v2f bf;
      bf.x = sW2[kb * O2 + col];
      bf.y = sW2[(kb + 1) * O2 + col];
      acc0 = wmma4f32(a0, bf, acc0);
      acc1 = wmma4f32(a1, bf, acc1);
    }
    const float s = sS2[col], sh = sB2[col];
    float mx = -3.0e38f;
#pragma unroll
    for (int v = 0; v < 8; ++v) {
      float h0 = acc0[v] * s + sh; h0 = (h0 > 0.f) ? h0 : 0.2f * h0;
      float h1 = acc1[v] * s + sh; h1 = (h1 > 0.f) ? h1 : 0.2f * h1;
      mx = fmaxf(mx, fmaxf(h0, h1));
    }
    mx = fmaxf(mx, __shfl_xor(mx, 16, 32));   // merge complementary row halves
    if (lane < 16) fout[(size_t)pt * O2 + col] = mx;
  }
}

// ---------------------------------------------------------------------------
// out (B,128,N) = transpose(concat([f1,f2], -1))
// ---------------------------------------------------------------------------
__global__ void concat_xpose_kernel(const float* __restrict__ f1,
                                    const float* __restrict__ f2,
                                    float* __restrict__ out) {
  const int i = blockIdx.x * blockDim.x + threadIdx.x;   // B_*128*N_
  const int n = i % N_;
  const int c = (i / N_) % 128;
  const int b = i / (N_ * 128);
  out[i] = (c < 64) ? f1[((size_t)b * N_ + n) * 64 + c]
                    : f2[((size_t)b * N_ + n) * 64 + (c - 64)];
}

// ---------------------------------------------------------------------------
extern "C" void kernel_launch(void* const* d_in, const int* in_sizes, int n_in,
                              void* d_out, int out_size, void* d_ws, size_t ws_size,
                              hipStream_t stream) {
  const float* x  = (const float*)d_in[0];
  const float* w1 = (const float*)d_in[1];
  const float* g1 = (const float*)d_in[2];
  const float* b1 = (const float*)d_in[3];
  const float* m1 = (const float*)d_in[4];
  const float* v1 = (const float*)d_in[5];
  const float* w2 = (const float*)d_in[6];
  const float* g2 = (const float*)d_in[7];
  const float* b2 = (const float*)d_in[8];
  const float* m2 = (const float*)d_in[9];
  const float* v2 = (const float*)d_in[10];
  const float* w3 = (const float*)d_in[11];
  const float* g3 = (const float*)d_in[12];
  const float* b3 = (const float*)d_in[13];
  const float* m3 = (const float*)d_in[14];
  const float* v3 = (const float*)d_in[15];
  const float* w4 = (const float*)d_in[16];
  const float* g4 = (const float*)d_in[17];
  const float* b4 = (const float*)d_in[18];
  const float* m4 = (const float*)d_in[19];
  const float* v4 = (const float*)d_in[20];

  float* p   = (float*)d_ws;              // B*N*3
  float* f1  = p  + (size_t)B_ * N_ * 3;  // B*N*64
  float* f2  = f1 + (size_t)B_ * N_ * 64; // B*N*64
  float* sqn = f2 + (size_t)B_ * N_ * 64; // B*N
  int* idx1  = (int*)(sqn + (size_t)B_ * N_);  // B*N*32
  int* idx2  = idx1 + (size_t)B_ * N_ * KNN;   // B*N*32

  constexpr int NW = 4;

  xpose_kernel<<<B_ * N_ / 256, 256, 0, stream>>>(x, p);
  topk_xyz_kernel<<<B_ * N_ / 128, 128, 0, stream>>>(p, idx1);

  {
    const int smem = (8 * 128 + 128 * 64 + 2 * 128 + 2 * 64 + NW * 32 * (8 + 128)) * 4;
    (void)hipFuncSetAttribute(reinterpret_cast<const void*>(edge_mlp_kernel<3, 8, 128, 64>),
                              hipFuncAttributeMaxDynamicSharedMemorySize, smem);
    edge_mlp_kernel<3, 8, 128, 64><<<B_ * N_ / NW, 128, smem, stream>>>(
        p, idx1, w1, g1, b1, m1, v1, w2, g2, b2, m2, v2, f1);
  }

  sqnorm64_kernel<<<B_ * N_ / 256, 256, 0, stream>>>(f1, sqn);
  topk_feat_kernel<<<B_ * N_ / 128, 128, 0, stream>>>(f1, sqn, idx2);

  {
    const int smem = (128 * 128 + 128 * 64 + 2 * 128 + 2 * 64 + NW * 32 * (128 + 128)) * 4;
    (void)hipFuncSetAttribute(reinterpret_cast<const void*>(edge_mlp_kernel<64, 128, 128, 64>),
                              hipFuncAttributeMaxDynamicSharedMemorySize, smem);
    edge_mlp_kernel<64, 128, 128, 64><<<B_ * N_ / NW, 128, smem, stream>>>(
        f1, idx2, w3, g3, b3, m3, v3, w4, g4, b4, m4, v4, f2);
  }

  concat_xpose_kernel<<<(B_ * 128 * N_) / 256, 256, 0, stream>>>(f1, f2, (float*)d_out);
}